// LocalRelationalLayer_52055003628042
// MI455X (gfx1250) — compile-verified
//
#include <hip/hip_runtime.h>
#include <hip/hip_bf16.h>

typedef __attribute__((ext_vector_type(2))) float v2f;
typedef __attribute__((ext_vector_type(8))) float v8f;

#define LL   4096   // H*W
#define CC   448    // channels
#define CGC  64     // key/query channels
#define K2C  49     // window size
#define GPH  32     // geometry-prior hidden
#define KB   16     // k-values staged in LDS per stage
#define BSTR 72     // padded LDS row stride (floats) -> conflict-free half-wave reads

// ---------------------------------------------------------------------------
// Geometry prior: gpk[g][p] = w2[g]·relu(w1·pos(p) + b1) + b2[g]
// pos(p) = (kj-3, 3-ki) with p = ki*7 + kj
// ---------------------------------------------------------------------------
__global__ void gp_kernel(const float* __restrict__ w1, const float* __restrict__ b1,
                          const float* __restrict__ w2, const float* __restrict__ b2,
                          float* __restrict__ gpk) {
    int idx = blockIdx.x * blockDim.x + threadIdx.x;
    if (idx >= CGC * K2C) return;
    int g = idx / K2C, p = idx % K2C;
    int ki = p / 7, kj = p % 7;
    float xp = (float)(kj - 3);
    float yp = (float)(3 - ki);
    float acc = b2[g];
    #pragma unroll
    for (int c = 0; c < GPH; ++c) {
        float h = w1[2 * c + 0] * xp + w1[2 * c + 1] * yp + b1[c];
        h = h > 0.f ? h : 0.f;
        acc += w2[g * GPH + c] * h;
    }
    gpk[idx] = acc;
}

// ---------------------------------------------------------------------------
// D[Mrows x 4096] = W[Mrows x 448] * X[448 x 4096] + bias, f32 WMMA 16x16x4.
// Block = 128 threads (4 waves). Block tile: 64 rows x 64 cols.
//   - B panel (KB x 64) staged in LDS once per k-stage, shared by all 4 waves
//     (4x reduction in L2 B-traffic vs per-wave loads).
//   - Each wave owns a 16-row slice; A fragments loaded direct from global.
// grid.x = Mrows/64, grid.y = 4096/64.
// ---------------------------------------------------------------------------
__global__ void gemm_wmma_f32(const float* __restrict__ W, const float* __restrict__ bias,
                              const float* __restrict__ X, float* __restrict__ D) {
    __shared__ float Bs[KB * BSTR];

    const int tid  = threadIdx.x;
    const int lane = tid & 31;
    const int wave = tid >> 5;
    const int lo   = lane & 15;     // M index (A) / N index (B,D) within half-wave
    const int hi   = lane >> 4;     // K-pair half (A,B) / M-half (D)
    const int row0 = blockIdx.x * 64 + wave * 16;
    const int col0 = blockIdx.y * 64;

    v8f acc0 = {}, acc1 = {}, acc2 = {}, acc3 = {};

    // A[lo][k + 2*hi + v]  (row-major W, row stride CC)
    const float* wrow  = W + (size_t)(row0 + lo) * CC + 2 * hi;
    const float* xbase = X + col0;

    // stage-load mapping: thread -> (row, 8-float chunk), 2x b128 per thread
    const int srow = tid >> 3;        // 0..15
    const int soff = (tid & 7) * 8;   // 0..56

    for (int k0 = 0; k0 < CC; k0 += KB) {
        // ---- cooperative stage of B[k0..k0+KB)[col0..col0+64) into LDS ----
        {
            const float4* src = (const float4*)(xbase + (size_t)(k0 + srow) * LL + soff);
            float4 p0 = src[0];
            float4 p1 = src[1];
            float4* dst = (float4*)(Bs + srow * BSTR + soff);
            dst[0] = p0;
            dst[1] = p1;
        }
        __syncthreads();

        // ---- 4 WMMA k-substeps from LDS ----
        #pragma unroll
        for (int ks = 0; ks < KB; ks += 4) {
            const int k = k0 + ks;
            v2f a;
            a.x = wrow[k];
            a.y = wrow[k + 1];
            const float* bs = Bs + (ks + 2 * hi) * BSTR + lo;
            v2f b0, b1, b2, b3;
            b0.x = bs[0];       b0.y = bs[BSTR];
            b1.x = bs[16];      b1.y = bs[BSTR + 16];
            b2.x = bs[32];      b2.y = bs[BSTR + 32];
            b3.x = bs[48];      b3.y = bs[BSTR + 48];
            acc0 = __builtin_amdgcn_wmma_f32_16x16x4_f32(false, a, false, b0, (short)0, acc0, false, false);
            acc1 = __builtin_amdgcn_wmma_f32_16x16x4_f32(false, a, false, b1, (short)0, acc1, false, false);
            acc2 = __builtin_amdgcn_wmma_f32_16x16x4_f32(false, a, false, b2, (short)0, acc2, false, false);
            acc3 = __builtin_amdgcn_wmma_f32_16x16x4_f32(false, a, false, b3, (short)0, acc3, false, false);
        }
        __syncthreads();
    }

    // D layout: VGPR r holds D[row0 + r + 8*hi][col0 + lo (+16t)]
    #pragma unroll
    for (int r = 0; r < 8; ++r) {
        int row = row0 + r + 8 * hi;
        float bv = bias[row];
        float* drow = D + (size_t)row * LL + col0 + lo;
        drow[0]  = acc0[r] + bv;
        drow[16] = acc1[r] + bv;
        drow[32] = acc2[r] + bv;
        drow[48] = acc3[r] + bv;
    }
}

// ---------------------------------------------------------------------------
// Attention: per (g, l): a[p] = km[g, l+dp]*qm[g,l] + gpk[g,p] (0 outside -> gpk),
// softmax over 49, then pre[m*64+g, l] = sum_p ck[p] * x[m*64+g, l+dp].
// grid = (64 g, 16), block = 256 -> l coalesced over j.
// ---------------------------------------------------------------------------
__global__ void attn_kernel(const float* __restrict__ km, const float* __restrict__ qm,
                            const float* __restrict__ x, const float* __restrict__ gpk,
                            float* __restrict__ pre) {
    const int g = blockIdx.x;
    const int l = blockIdx.y * blockDim.x + threadIdx.x;
    const int i = l >> 6;
    const int j = l & 63;

    const float q = qm[(size_t)g * LL + l];

    float e[K2C];
    float mx = -3.402823466e38f;
    #pragma unroll
    for (int p = 0; p < K2C; ++p) {
        int ki = p / 7, kj = p % 7;
        int ii = i + ki - 3, jj = j + kj - 3;
        bool ok = ((unsigned)ii < 64u) && ((unsigned)jj < 64u);
        float kv = ok ? km[(size_t)g * LL + ii * 64 + jj] : 0.f;
        float a = kv * q + gpk[g * K2C + p];
        e[p] = a;
        mx = fmaxf(mx, a);
    }
    float s = 0.f;
    #pragma unroll
    for (int p = 0; p < K2C; ++p) {
        float ev = expf(e[p] - mx);
        e[p] = ev;
        s += ev;
    }
    const float inv = 1.f / s;

    for (int m = 0; m < 7; ++m) {
        const int c = m * CGC + g;
        const float* xc = x + (size_t)c * LL;
        float acc = 0.f;
        #pragma unroll
        for (int p = 0; p < K2C; ++p) {
            int ki = p / 7, kj = p % 7;
            int ii = i + ki - 3, jj = j + kj - 3;
            bool ok = ((unsigned)ii < 64u) && ((unsigned)jj < 64u);
            float xv = ok ? xc[ii * 64 + jj] : 0.f;
            acc += e[p] * xv;
        }
        pre[(size_t)c * LL + l] = acc * inv;
    }
}

// ---------------------------------------------------------------------------
// Host-side launch. Inputs (setup_inputs order):
// 0:x 1:Wk 2:bk 3:Wq 4:bq 5:gp_w1 6:gp_b1 7:gp_w2 8:gp_b2 9:Wf 10:bf
// Workspace: gpk (3136) | km (64*4096) | qm (64*4096) | pre (448*4096) floats
// ---------------------------------------------------------------------------
extern "C" void kernel_launch(void* const* d_in, const int* in_sizes, int n_in,
                              void* d_out, int out_size, void* d_ws, size_t ws_size,
                              hipStream_t stream) {
    const float* x   = (const float*)d_in[0];
    const float* Wk  = (const float*)d_in[1];
    const float* bk  = (const float*)d_in[2];
    const float* Wq  = (const float*)d_in[3];
    const float* bq  = (const float*)d_in[4];
    const float* gw1 = (const float*)d_in[5];
    const float* gb1 = (const float*)d_in[6];
    const float* gw2 = (const float*)d_in[7];
    const float* gb2 = (const float*)d_in[8];
    const float* Wf  = (const float*)d_in[9];
    const float* bf  = (const float*)d_in[10];
    float* out = (float*)d_out;

    float* gpk = (float*)d_ws;
    float* km  = gpk + CGC * K2C;
    float* qm  = km + (size_t)CGC * LL;
    float* pre = qm + (size_t)CGC * LL;

    // 1) geometry prior (64*49 values)
    gp_kernel<<<dim3(13), dim3(256), 0, stream>>>(gw1, gb1, gw2, gb2, gpk);
    // 2) key / query 1x1 convs: 64x448 * 448x4096 (LDS-staged B panel)
    gemm_wmma_f32<<<dim3(1, 64), dim3(128), 0, stream>>>(Wk, bk, x, km);
    gemm_wmma_f32<<<dim3(1, 64), dim3(128), 0, stream>>>(Wq, bq, x, qm);
    // 3) local-relation attention + aggregation
    attn_kernel<<<dim3(64, 16), dim3(256), 0, stream>>>(km, qm, x, gpk, pre);
    // 4) final 1x1 conv: 448x448 * 448x4096 (LDS-staged B panel)
    gemm_wmma_f32<<<dim3(7, 64), dim3(128), 0, stream>>>(Wf, bf, pre, out);
}